// BERT_18494129177014
// MI455X (gfx1250) — compile-verified
//
#include <hip/hip_runtime.h>
#include <hip/hip_bf16.h>

typedef __attribute__((ext_vector_type(2))) float v2f;
typedef __attribute__((ext_vector_type(8))) float v8f;

#define BB   128
#define LQ   128
#define LD   512
#define DD   768
#define NIT  500
#define SREG 1.0f
#define SEPS 1e-12f

// ---------------------------------------------------------------------------
// Kernel 1: per-row inverse L2 norms.  rq[B*LQ] then rd[B*LD] in workspace.
// One wave (32 lanes) per row; 768/32 = 24 elements per lane.
// ---------------------------------------------------------------------------
__global__ __launch_bounds__(256) void k_invnorm(const float* __restrict__ q,
                                                 const float* __restrict__ d,
                                                 float* __restrict__ rq,
                                                 float* __restrict__ rd) {
    const int lane = threadIdx.x & 31;
    const int row  = (blockIdx.x * (blockDim.x >> 5)) + (threadIdx.x >> 5);
    const int nq   = BB * LQ;
    const float* src;
    float* dst;
    if (row < nq) { src = q + (size_t)row * DD;        dst = rq + row; }
    else          { src = d + (size_t)(row - nq) * DD; dst = rd + (row - nq); }
    float acc = 0.f;
    #pragma unroll
    for (int i = 0; i < DD / 32; ++i) {
        float x = src[lane + i * 32];
        acc = fmaf(x, x, acc);
    }
    #pragma unroll
    for (int off = 16; off > 0; off >>= 1) acc += __shfl_xor(acc, off, 32);
    if (lane == 0) *dst = 1.0f / fmaxf(sqrtf(acc), SEPS);
}

// ---------------------------------------------------------------------------
// Kernel 2: C = 1 - (q . d) * rq * rd   via V_WMMA_F32_16X16X4_F32.
// Block = 256 threads = 8 waves; each wave owns one 16x16 C tile.
// blockIdx.x = ((b*8 + rowTile)*4 + colGroup); wave w -> colTile = cg*8 + w.
// A fragment (16x4, f32): lane l -> A[l%16][k0 + 2*(l/16) + v], v=0,1
// B fragment (4x16, f32): lane l -> dn[l%16][k0 + 2*(l/16) + v]  (B = d^T)
// ---------------------------------------------------------------------------
__global__ __launch_bounds__(256) void k_cost(const float* __restrict__ q,
                                              const float* __restrict__ d,
                                              const float* __restrict__ rq,
                                              const float* __restrict__ rd,
                                              float* __restrict__ Cout) {
    const int wid  = threadIdx.x >> 5;
    const int lane = threadIdx.x & 31;
    const int cg   = blockIdx.x & 3;
    const int rt   = (blockIdx.x >> 2) & 7;
    const int b    = blockIdx.x >> 5;
    const int ct   = cg * 8 + wid;

    const int m = lane & 15;      // row within A-tile / col within B-tile
    const int h = lane >> 4;      // selects k pair

    const float* arow = q + ((size_t)(b * LQ + rt * 16 + m)) * DD + 2 * h;
    const float* brow = d + ((size_t)(b * LD + ct * 16 + m)) * DD + 2 * h;

    v8f acc = {0.f, 0.f, 0.f, 0.f, 0.f, 0.f, 0.f, 0.f};
    #pragma unroll 4
    for (int k0 = 0; k0 < DD; k0 += 4) {
        v2f a = *(const v2f*)(arow + k0);
        v2f bv = *(const v2f*)(brow + k0);
        acc = __builtin_amdgcn_wmma_f32_16x16x4_f32(
            false, a, false, bv, (short)0, acc, false, false);
    }

    const int colg = ct * 16 + m;
    const float rdv = rd[b * LD + colg];
    #pragma unroll
    for (int j = 0; j < 8; ++j) {
        const int rowg = rt * 16 + j + 8 * h;   // D layout: VGPR j -> M=j / M=j+8
        const float rqv = rq[b * LQ + rowg];
        Cout[((size_t)(b * LQ + rowg)) * LD + colg] = 1.0f - acc[j] * rqv * rdv;
    }
}

// ---------------------------------------------------------------------------
// Kernel 3: Sinkhorn, one workgroup (1024 threads, 32 waves) per batch.
// K = exp(-C/reg) lives in LDS (256 KB) for all 500 iterations.
// ---------------------------------------------------------------------------
__global__ __launch_bounds__(1024) void k_sinkhorn(const float* __restrict__ Cin,
                                                   const int* __restrict__ qmask,
                                                   const int* __restrict__ dmask,
                                                   float* __restrict__ dist,
                                                   float* __restrict__ Tout) {
    extern __shared__ float smem[];
    float* Ks  = smem;                 // LQ*LD = 65536
    float* us  = Ks + LQ * LD;         // 128
    float* vs  = us + LQ;              // 512
    float* qd  = vs + LD;              // 128
    float* dd  = qd + LQ;              // 512
    float* red = dd + LD;              // 1024

    const int tid = threadIdx.x;
    const int b   = blockIdx.x;
    const float* Cb = Cin + (size_t)b * (LQ * LD);

    // masks -> LDS
    if (tid < LQ) qd[tid] = (float)qmask[b * LQ + tid];
    if (tid < LD) dd[tid] = (float)dmask[b * LD + tid];
    __syncthreads();

    // normalize mask distributions (redundant broadcast sums: cheap, no races)
    float sq = 0.f, sd = 0.f;
    for (int i = 0; i < LQ; ++i) sq += qd[i];
    for (int i = 0; i < LD; ++i) sd += dd[i];
    __syncthreads();
    if (tid < LQ) { qd[tid] = qd[tid] / sq; us[tid] = 1.0f; }
    if (tid < LD) dd[tid] = dd[tid] / sd;

    // K = exp(-C/reg) into LDS, coalesced
    for (int i = tid; i < LQ * LD; i += 1024)
        Ks[i] = __expf(-Cb[i] * (1.0f / SREG));
    __syncthreads();

    const int w    = tid >> 5;
    const int lane = tid & 31;
    const int col  = tid & 511;
    const int half = tid >> 9;          // split q-range over the two thread halves

    for (int it = 0; it < NIT; ++it) {
        // v = d_dist / (K^T u): 1024 threads, each half sums 64 q's of one column
        {
            float acc = 0.f;
            const int q0 = half * (LQ / 2);
            #pragma unroll 8
            for (int qq = 0; qq < LQ / 2; ++qq)
                acc = fmaf(Ks[(q0 + qq) * LD + col], us[q0 + qq], acc);
            red[tid] = acc;
        }
        __syncthreads();
        if (tid < LD) vs[tid] = dd[tid] / (red[tid] + red[tid + LD]);
        __syncthreads();

        // u = q_dist / (K v): wave w handles rows 4w..4w+3, lane-strided cols
        #pragma unroll
        for (int r = 0; r < 4; ++r) {
            const int qq = w * 4 + r;
            float acc = 0.f;
            #pragma unroll
            for (int k = lane; k < LD; k += 32)
                acc = fmaf(Ks[qq * LD + k], vs[k], acc);
            #pragma unroll
            for (int off = 16; off > 0; off >>= 1) acc += __shfl_xor(acc, off, 32);
            if (lane == 0) us[qq] = qd[qq] / acc;
        }
        __syncthreads();
    }

    // T = u (*) K (*) v ; distance = sum(C * T)
    float dpart = 0.f;
    float* Tb = Tout + (size_t)b * (LQ * LD);
    for (int i = tid; i < LQ * LD; i += 1024) {
        const int qq = i >> 9;
        const int kk = i & 511;
        const float t = us[qq] * Ks[i] * vs[kk];
        Tb[i] = t;
        dpart = fmaf(Cb[i], t, dpart);
    }
    red[tid] = dpart;
    __syncthreads();
    for (int s = 512; s > 0; s >>= 1) {
        if (tid < s) red[tid] += red[tid + s];
        __syncthreads();
    }
    if (tid == 0) dist[b] = red[0];
}

// ---------------------------------------------------------------------------
extern "C" void kernel_launch(void* const* d_in, const int* in_sizes, int n_in,
                              void* d_out, int out_size, void* d_ws, size_t ws_size,
                              hipStream_t stream) {
    const float* q     = (const float*)d_in[0];
    const float* d     = (const float*)d_in[1];
    const int*   qmask = (const int*)d_in[2];
    const int*   dmask = (const int*)d_in[3];

    float* out   = (float*)d_out;
    float* dist  = out;                          // [B]
    float* Cout  = out + BB;                     // [B*LQ*LD]
    float* Tout  = Cout + (size_t)BB * LQ * LD;  // [B*LQ*LD]

    float* rq = (float*)d_ws;                    // [B*LQ]
    float* rd = rq + BB * LQ;                    // [B*LD]

    // 1) inverse norms: one wave per row, 8 waves per block
    {
        const int rows = BB * LQ + BB * LD;      // 81920
        k_invnorm<<<rows / 8, 256, 0, stream>>>(q, d, rq, rd);
    }
    // 2) cost matrix GEMM via f32 WMMA
    {
        k_cost<<<BB * 8 * 4, 256, 0, stream>>>(q, d, rq, rd, Cout);
    }
    // 3) Sinkhorn: one workgroup per batch, K resident in LDS
    {
        const size_t smem = (size_t)(LQ * LD + LQ + LD + LQ + LD + 1024) * sizeof(float);
        k_sinkhorn<<<BB, 1024, smem, stream>>>(Cout, qmask, dmask, dist, Tout);
    }
    (void)in_sizes; (void)n_in; (void)out_size; (void)ws_size;
}